// Chamfer_Loss_70300024701626
// MI455X (gfx1250) — compile-verified
//
#include <hip/hip_runtime.h>

typedef __attribute__((ext_vector_type(2))) float v2f;
typedef __attribute__((ext_vector_type(8))) float v8f;

#define TILE_PTS 1024   // loop-cloud points staged in LDS per chunk (16 KB as 2x v2f)
#define NPTS     4096
#define BATCH    16

// One block = one batch slice: 8 waves x 16 fixed points = 128 points of cloudA.
// Role swap vs naive mapping: the LOOPED cloud (cloudB) is the A matrix, the
// wave-fixed 16 cloudA points are the B matrix, so C (= ||x||^2 per column) is
// loop-invariant and lives in 8 VGPRs for the whole kernel.
//   A row  = (-2y0, -2y1, -2y2, ||y||^2)   (prebuilt in LDS, per-half float2)
//   B col  = ( x0,   x1,   x2,  1      )   (built once per wave)
//   C      = ||x||^2 broadcast per column  (fixed)
// => D tile = squared-distance tile [16 loop pts x 16 fixed pts].
// Min over the loop dim = min over ROWS: per lane that's a tree-min of its 8
// D values into a single scalar accumulator; one shfl_xor(16) at the end
// merges the two row halves.
__global__ __launch_bounds__(256)
void chamfer_min_kernel(const float* __restrict__ cloudA,   // [B, N, 3] fixed (B-side)
                        const float* __restrict__ cloudB,   // [B, M, 3] looped (A-side)
                        float* __restrict__ minOut,         // [B, N]
                        int N, int M)
{
    __shared__ v2f sm[2 * TILE_PTS];   // [0..1023]: (-2y0,-2y1)  [1024..]: (-2y2,||y||^2)

    const int b       = blockIdx.y;
    const int lane    = threadIdx.x & 31;
    const int wave    = threadIdx.x >> 5;
    const int halfSel = lane >> 4;     // 0 -> K0/K1 half, 1 -> K2/K3 half
    const int l16     = lane & 15;

    // ---- Fixed B fragment + C fragment for this wave's 16 points ----------
    const int    colBase = blockIdx.x * 128 + wave * 16;
    const float* ap      = cloudA + ((size_t)b * N + colBase + l16) * 3;
    const float  x0 = ap[0], x1 = ap[1], x2 = ap[2];
    const float  sqx = x0 * x0 + x1 * x1 + x2 * x2;

    v2f bfrag;                          // lanes 0-15: (K0,K1)=(x0,x1); 16-31: (K2,K3)=(x2,1)
    bfrag.x = halfSel ? x2 : x0;
    bfrag.y = halfSel ? 1.0f : x1;

    v8f cfrag;                          // C[m][n] = ||x_n||^2  (column = lane%16)
#pragma unroll
    for (int i = 0; i < 8; ++i) cfrag[i] = sqx;

    float acc = 3.0e38f;                // running min over all loop points

    // Per-lane LDS base: pick the half-array once, address = base + t + l16.
    const v2f* smSel = sm + halfSel * TILE_PTS + l16;

    for (int chunk = 0; chunk < M; chunk += TILE_PTS) {
        __syncthreads();
        // Stage cloudB chunk as prebuilt A-fragment halves.
        for (int p = threadIdx.x; p < TILE_PTS; p += 256) {
            const float* bp = cloudB + ((size_t)b * M + chunk + p) * 3;
            const float y0 = bp[0], y1 = bp[1], y2 = bp[2];
            v2f lo, hi;
            lo.x = -2.0f * y0;
            lo.y = -2.0f * y1;
            hi.x = -2.0f * y2;
            hi.y = y0 * y0 + y1 * y1 + y2 * y2;
            sm[p]            = lo;
            sm[TILE_PTS + p] = hi;
        }
        __syncthreads();

#pragma unroll 8
        for (int t = 0; t < TILE_PTS; t += 16) {
            const v2f afrag = smSel[t];                 // single ds_load_b64

            const v8f d = __builtin_amdgcn_wmma_f32_16x16x4_f32(
                false, afrag, false, bfrag, (short)0, cfrag, false, false);

            // min over this tile's 8 rows (loop points) held by this lane
            const float m01 = fminf(fminf(d[0], d[1]), d[2]);
            const float m23 = fminf(fminf(d[3], d[4]), d[5]);
            const float m45 = fminf(d[6], d[7]);
            acc = fminf(acc, fminf(fminf(m01, m23), m45));
        }
    }

    // Merge row halves: lane L (rows 0-7 of col L) with lane L+16 (rows 8-15).
    acc = fminf(acc, __shfl_xor(acc, 16, 32));

    if (lane < 16) {
        minOut[(size_t)b * N + colBase + l16] = acc;
    }
}

__global__ __launch_bounds__(256)
void chamfer_reduce_kernel(const float* __restrict__ minA,
                           const float* __restrict__ minB,
                           int count, float* __restrict__ out)
{
    __shared__ float s1[256];
    __shared__ float s2[256];
    float a = 0.0f, c = 0.0f;
    for (int i = threadIdx.x; i < count; i += 256) {
        a += minA[i];
        c += minB[i];
    }
    s1[threadIdx.x] = a;
    s2[threadIdx.x] = c;
    __syncthreads();
    for (int off = 128; off > 0; off >>= 1) {
        if (threadIdx.x < off) {
            s1[threadIdx.x] += s1[threadIdx.x + off];
            s2[threadIdx.x] += s2[threadIdx.x + off];
        }
        __syncthreads();
    }
    if (threadIdx.x == 0) {
        const float inv = 1.0f / (float)count;
        out[0] = fmaxf(s1[0] * inv, s2[0] * inv);
    }
}

extern "C" void kernel_launch(void* const* d_in, const int* in_sizes, int n_in,
                              void* d_out, int out_size, void* d_ws, size_t ws_size,
                              hipStream_t stream) {
    const float* inputs = (const float*)d_in[0];  // [16, 4096, 3] f32
    const float* preds  = (const float*)d_in[1];  // [16, 4096, 3] f32

    float* minA = (float*)d_ws;                   // [16*4096] min over preds per input
    float* minB = minA + BATCH * NPTS;            // [16*4096] min over inputs per pred

    dim3 grid(NPTS / 128, BATCH);
    // Direction 1: inputs are the fixed/output points, preds are looped.
    chamfer_min_kernel<<<grid, 256, 0, stream>>>(inputs, preds, minA, NPTS, NPTS);
    // Direction 2: swap roles (N == M so same kernel).
    chamfer_min_kernel<<<grid, 256, 0, stream>>>(preds, inputs, minB, NPTS, NPTS);

    chamfer_reduce_kernel<<<1, 256, 0, stream>>>(minA, minB, BATCH * NPTS,
                                                 (float*)d_out);
}